// Net_multi_11390253269716
// MI455X (gfx1250) — compile-verified
//
#include <hip/hip_runtime.h>
#include <hip/hip_bf16.h>

// ---------------------------------------------------------------------------
// Graph U-Net (GCN) on a 784x480 grid, C=32 channels, for MI455X (gfx1250).
//  - Dense [N,32]@[32,32] linears via V_WMMA_F32_16X16X4_F32 (wave32, one wave
//    per 16x32 output tile). W staged TRANSPOSED in LDS with pad-34 so every
//    B fragment is one conflict-free, 8B-aligned ds_load_b64 (no repacking).
//  - Edge scatter: one wave per edge, lane == channel (wave32 == C=32),
//    global_atomic_add_f32 into L2 (48MB feature maps fit in 192MB L2).
// ---------------------------------------------------------------------------

#define NX 784
#define NY 480
#define N0 (NX * NY)      // 376320
#define N1 (N0 / 4)       // 94080
#define N2 (N0 / 16)      // 23520
#define E0 (4 * N0)
#define E1 (4 * N1)
#define E2 (4 * N2)

typedef __attribute__((ext_vector_type(2))) float v2f;
typedef __attribute__((ext_vector_type(8))) float v8f;

// ---------------- degree / normalization ----------------------------------

__global__ void deg_init_kernel(float* deg, int n) {
    int i = blockIdx.x * blockDim.x + threadIdx.x;
    if (i < n) deg[i] = 1.0f;                     // self loop contributes 1
}

__global__ void deg_accum_kernel(const int* __restrict__ dst, float* deg, int E) {
    int e = blockIdx.x * blockDim.x + threadIdx.x;
    if (e < E) atomicAdd(&deg[dst[e]], 1.0f);
}

__global__ void deg_finalize_kernel(float* dinv, int n) {
    int i = blockIdx.x * blockDim.x + threadIdx.x;
    if (i < n) dinv[i] = rsqrtf(dinv[i]);         // deg >= 1 always (self loop)
}

// ---------------- fc1: [N,4] @ [4,32] + b, ReLU ----------------------------

__global__ void fc1_relu_kernel(const float* __restrict__ x,
                                const float* __restrict__ w,
                                const float* __restrict__ b,
                                float* __restrict__ h, int n) {
    int t = blockIdx.x * blockDim.x + threadIdx.x;
    if (t >= n * 32) return;
    int i = t >> 5;
    int c = t & 31;
    float s = b[c];
    const float* xp = x + (size_t)i * 4;
    s += xp[0] * w[0 * 32 + c];
    s += xp[1] * w[1 * 32 + c];
    s += xp[2] * w[2 * 32 + c];
    s += xp[3] * w[3 * 32 + c];
    h[(size_t)i * 32 + c] = fmaxf(s, 0.0f);
}

// ---------------- dense [N,32] @ [32,32] via WMMA f32 16x16x4 --------------
// One wave computes a 16x32 output tile. A-matrix VGPR layout (ISA 7.12.2):
//   lane = khalf*16 + m holds A[m][k0+2*khalf] (v0) and A[m][k0+2*khalf+1] (v1)
// B mirrored with lane = khalf*16 + n. D: VGPR j -> row j (lanes 0-15) /
// row j+8 (lanes 16-31), col = lane&15.
// LDS holds W^T with row pitch 34 floats: even pitch -> 8B-aligned b64 loads;
// (34*n + k) mod 64 distinct even banks for n=0..15, and the +2 offset of the
// upper half-wave never aliases (17 invertible mod 32) -> conflict-free.

#define WT_PITCH 34

__global__ void __launch_bounds__(256)
linear32_wmma_kernel(const float* __restrict__ X, const float* __restrict__ W,
                     float* __restrict__ Y, int n) {
    __shared__ float wldsT[32 * WT_PITCH];
    for (int i = threadIdx.x; i < 1024; i += blockDim.x) {
        int k = i >> 5;          // row of W
        int c = i & 31;          // col of W
        wldsT[c * WT_PITCH + k] = W[i];
    }
    __syncthreads();

    int lane  = threadIdx.x & 31;
    int wave  = threadIdx.x >> 5;
    int tile  = blockIdx.x * (blockDim.x >> 5) + wave;   // 16-row tile
    int tiles = n >> 4;                                   // n is a multiple of 16
    if (tile >= tiles) return;                            // uniform wave exit

    int row0 = tile << 4;
    int m    = lane & 15;
    int kh   = lane >> 4;
    const float* arow = X + (size_t)(row0 + m) * 32;
    const float* bcol0 = wldsT + m * WT_PITCH;            // cols 0..15
    const float* bcol1 = wldsT + (m + 16) * WT_PITCH;     // cols 16..31

    v8f acc0 = {};
    v8f acc1 = {};
#pragma unroll
    for (int k0 = 0; k0 < 32; k0 += 4) {
        int krow = k0 + 2 * kh;
        v2f a  = *(const v2f*)(arow + krow);              // global, 8B aligned
        v2f b0 = *(const v2f*)(bcol0 + krow);             // one ds_load_b64
        v2f b1 = *(const v2f*)(bcol1 + krow);             // one ds_load_b64
        acc0 = __builtin_amdgcn_wmma_f32_16x16x4_f32(false, a, false, b0,
                                                     (short)0, acc0, false, false);
        acc1 = __builtin_amdgcn_wmma_f32_16x16x4_f32(false, a, false, b1,
                                                     (short)0, acc1, false, false);
    }

#pragma unroll
    for (int j = 0; j < 8; ++j) {
        int rr = row0 + j + 8 * kh;
        Y[(size_t)rr * 32 + m]      = acc0[j];
        Y[(size_t)rr * 32 + 16 + m] = acc1[j];
    }
}

// ---------------- GCN scatter ----------------------------------------------

__global__ void init_bias_kernel(float* __restrict__ y, const float* __restrict__ b,
                                 int n) {
    int t = blockIdx.x * blockDim.x + threadIdx.x;
    if (t >= n * 32) return;
    y[t] = b[t & 31];
}

// one wave per edge, lane == channel: fully coalesced 128B gather + atomic
__global__ void gcn_scatter_kernel(const int* __restrict__ src,
                                   const int* __restrict__ dst,
                                   const float* __restrict__ xw,
                                   const float* __restrict__ dinv,
                                   float* __restrict__ y, int E) {
    int e = blockIdx.x * (blockDim.x >> 5) + (threadIdx.x >> 5);
    if (e >= E) return;
    int c = threadIdx.x & 31;
    int s = src[e];
    int d = dst[e];
    float nrm = dinv[s] * dinv[d];
    atomicAdd(&y[(size_t)d * 32 + c], xw[(size_t)s * 32 + c] * nrm);
}

// self-loop contribution + ReLU (separate launch -> no race with edge atomics)
__global__ void gcn_self_relu_kernel(const float* __restrict__ xw,
                                     const float* __restrict__ dinv,
                                     float* __restrict__ y, int n) {
    int t = blockIdx.x * blockDim.x + threadIdx.x;
    if (t >= n * 32) return;
    int i = t >> 5;
    float di = dinv[i];
    float v = y[t] + xw[t] * di * di;
    y[t] = fmaxf(v, 0.0f);
}

// ---------------- resampling -----------------------------------------------

// out grid Ho x Wo, input grid (2Ho) x (2Wo): out[r][c] = in[2r][2c]
__global__ void downsample_kernel(const float* __restrict__ in,
                                  float* __restrict__ out, int Ho, int Wo) {
    int t = blockIdx.x * blockDim.x + threadIdx.x;
    if (t >= Ho * Wo * 32) return;
    int c  = t & 31;
    int nd = t >> 5;
    int r  = nd / Wo;
    int cc = nd - r * Wo;
    int src = (2 * r) * (2 * Wo) + 2 * cc;
    out[t] = in[(size_t)src * 32 + c];
}

// out grid H x W: out = base + up2(small), small grid (H/2) x (W/2)
__global__ void upsample_add_kernel(const float* __restrict__ base,
                                    const float* __restrict__ small,
                                    float* __restrict__ out, int H, int W) {
    int t = blockIdx.x * blockDim.x + threadIdx.x;
    if (t >= H * W * 32) return;
    int c  = t & 31;
    int nd = t >> 5;
    int r  = nd / W;
    int cc = nd - r * W;
    int p  = (r >> 1) * (W >> 1) + (cc >> 1);
    out[t] = base[t] + small[(size_t)p * 32 + c];
}

// ---------------- fc2: [N,32] @ [32,3] + b ---------------------------------

__global__ void fc2_kernel(const float* __restrict__ h, const float* __restrict__ w,
                           const float* __restrict__ b, float* __restrict__ out,
                           int n) {
    int i = blockIdx.x * blockDim.x + threadIdx.x;
    if (i >= n) return;
    float s0 = b[0], s1 = b[1], s2 = b[2];
    const float* hp = h + (size_t)i * 32;
#pragma unroll
    for (int k = 0; k < 32; ++k) {
        float v = hp[k];
        s0 += v * w[k * 3 + 0];
        s1 += v * w[k * 3 + 1];
        s2 += v * w[k * 3 + 2];
    }
    out[(size_t)i * 3 + 0] = s0;
    out[(size_t)i * 3 + 1] = s1;
    out[(size_t)i * 3 + 2] = s2;
}

// ---------------------------------------------------------------------------

static void run_gcn(float* xio,            // in/out feature buffer [n,32]
                    float* xw,             // scratch for X@W
                    const float* w, const float* b,
                    const int* src, const int* dst, const float* dinv,
                    int n, int E, hipStream_t stream) {
    int tiles = n >> 4;
    linear32_wmma_kernel<<<(tiles + 7) / 8, 256, 0, stream>>>(xio, w, xw, n);
    init_bias_kernel<<<(n * 32 + 255) / 256, 256, 0, stream>>>(xio, b, n);
    gcn_scatter_kernel<<<(E + 7) / 8, 256, 0, stream>>>(src, dst, xw, dinv, xio, E);
    gcn_self_relu_kernel<<<(n * 32 + 255) / 256, 256, 0, stream>>>(xw, dinv, xio, n);
}

extern "C" void kernel_launch(void* const* d_in, const int* in_sizes, int n_in,
                              void* d_out, int out_size, void* d_ws, size_t ws_size,
                              hipStream_t stream) {
    (void)in_sizes; (void)n_in; (void)out_size; (void)ws_size;

    const float* x     = (const float*)d_in[0];
    const float* fc1_w = (const float*)d_in[1];
    const float* fc1_b = (const float*)d_in[2];
    const float* w1    = (const float*)d_in[3];
    const float* b1    = (const float*)d_in[4];
    const float* w2    = (const float*)d_in[5];
    const float* b2    = (const float*)d_in[6];
    const float* w3    = (const float*)d_in[7];
    const float* b3    = (const float*)d_in[8];
    const float* w4    = (const float*)d_in[9];
    const float* b4    = (const float*)d_in[10];
    const float* w5    = (const float*)d_in[11];
    const float* b5    = (const float*)d_in[12];
    const float* fc2_w = (const float*)d_in[13];
    const float* fc2_b = (const float*)d_in[14];
    // d_in[15..17] = idx0/idx1/idx2 (arange -> identity, unused)
    const int* ei0 = (const int*)d_in[18];
    const int* ei1 = (const int*)d_in[19];
    const int* ei2 = (const int*)d_in[20];
    const int *src0 = ei0, *dst0 = ei0 + E0;
    const int *src1 = ei1, *dst1 = ei1 + E1;
    const int *src2 = ei2, *dst2 = ei2 + E2;

    // ---- workspace carve-up (floats), 256B aligned chunks ----
    char* ws = (char*)d_ws;
    size_t off = 0;
    auto alloc = [&](size_t nfloats) {
        float* p = (float*)(ws + off);
        off += ((nfloats * sizeof(float)) + 255) & ~(size_t)255;
        return p;
    };
    float* dinv0  = alloc(N0);
    float* dinv1  = alloc(N1);
    float* dinv2  = alloc(N2);
    float* bufA   = alloc((size_t)N0 * 32);   // h (level 0)
    float* bufXW  = alloc((size_t)N0 * 32);   // shared X@W scratch (max size)
    float* bufHU  = alloc((size_t)N0 * 32);   // hu
    float* bufH1  = alloc((size_t)N1 * 32);   // h1
    float* bufH1U = alloc((size_t)N1 * 32);   // h1u
    float* bufH2  = alloc((size_t)N2 * 32);   // h2

    // ---- degree -> 1/sqrt(deg) per level (recomputed every call) ----
    deg_init_kernel<<<(N0 + 255) / 256, 256, 0, stream>>>(dinv0, N0);
    deg_accum_kernel<<<(E0 + 255) / 256, 256, 0, stream>>>(dst0, dinv0, E0);
    deg_finalize_kernel<<<(N0 + 255) / 256, 256, 0, stream>>>(dinv0, N0);

    deg_init_kernel<<<(N1 + 255) / 256, 256, 0, stream>>>(dinv1, N1);
    deg_accum_kernel<<<(E1 + 255) / 256, 256, 0, stream>>>(dst1, dinv1, E1);
    deg_finalize_kernel<<<(N1 + 255) / 256, 256, 0, stream>>>(dinv1, N1);

    deg_init_kernel<<<(N2 + 255) / 256, 256, 0, stream>>>(dinv2, N2);
    deg_accum_kernel<<<(E2 + 255) / 256, 256, 0, stream>>>(dst2, dinv2, E2);
    deg_finalize_kernel<<<(N2 + 255) / 256, 256, 0, stream>>>(dinv2, N2);

    // ---- encoder / decoder pipeline ----
    fc1_relu_kernel<<<((size_t)N0 * 32 + 255) / 256, 256, 0, stream>>>(
        x, fc1_w, fc1_b, bufA, N0);

    run_gcn(bufA, bufXW, w1, b1, src0, dst0, dinv0, N0, E0, stream);   // level 0

    downsample_kernel<<<((size_t)N1 * 32 + 255) / 256, 256, 0, stream>>>(
        bufA, bufH1, NX / 2, NY / 2);
    run_gcn(bufH1, bufXW, w2, b2, src1, dst1, dinv1, N1, E1, stream);  // level 1

    downsample_kernel<<<((size_t)N2 * 32 + 255) / 256, 256, 0, stream>>>(
        bufH1, bufH2, NX / 4, NY / 4);
    run_gcn(bufH2, bufXW, w3, b3, src2, dst2, dinv2, N2, E2, stream);  // level 2

    upsample_add_kernel<<<((size_t)N1 * 32 + 255) / 256, 256, 0, stream>>>(
        bufH1, bufH2, bufH1U, NX / 2, NY / 2);
    run_gcn(bufH1U, bufXW, w4, b4, src1, dst1, dinv1, N1, E1, stream); // level 1 up

    upsample_add_kernel<<<((size_t)N0 * 32 + 255) / 256, 256, 0, stream>>>(
        bufA, bufH1U, bufHU, NX, NY);
    run_gcn(bufHU, bufXW, w5, b5, src0, dst0, dinv0, N0, E0, stream);  // level 0 up

    fc2_kernel<<<(N0 + 255) / 256, 256, 0, stream>>>(
        bufHU, fc2_w, fc2_b, (float*)d_out, N0);
}